// MultiModalBrainGNN_4346506903772
// MI455X (gfx1250) — compile-verified
//
#include <hip/hip_runtime.h>
#include <hip/hip_bf16.h>

#define N_NODES 50000
#define N_EDGES 1600000
#define HID 256

typedef float v2f __attribute__((ext_vector_type(2)));
typedef float v8f __attribute__((ext_vector_type(8)));

// ---------------------------------------------------------------- init / degree
__global__ void init_ws_kernel(float4* __restrict__ agg4, float* __restrict__ deg,
                               float* __restrict__ acc, long total4) {
  long i = (long)blockIdx.x * blockDim.x + threadIdx.x;
  if (i < total4) agg4[i] = make_float4(0.f, 0.f, 0.f, 0.f);
  if (i < N_NODES) deg[i] = 1.0f;           // self-loop weight
  if (i == 0)      acc[0] = 0.0f;
}

__global__ void degree_kernel(const int* __restrict__ dst, float* __restrict__ deg) {
  int e = blockIdx.x * blockDim.x + threadIdx.x;
  if (e < N_EDGES) atomicAdd(&deg[dst[e]], 1.0f);
}

__global__ void rsqrt_kernel(const float* __restrict__ deg, float* __restrict__ dis) {
  int n = blockIdx.x * blockDim.x + threadIdx.x;
  if (n < N_NODES) dis[n] = rsqrtf(deg[n]);
}

// ---------------------------------------------------------------- WMMA f32 GEMM
// One wave computes a 16(M) x 64(N) tile of Out = A * B (+bias), K-step 4,
// using V_WMMA_F32_16X16X4_F32. Strides and K are compile-time so the backend
// folds them into immediate load offsets and unrolls the K loop.
// BT=true: B stored as [N,K] (W^T GEMM); pairs load as one b64.
// EPI=0: store (+optional bias). EPI=1: fused reduce: acc += sum_rows dot(row, outw).
template <bool BT, int EPI, int LDA, int LDB, int LDO, int K>
__global__ void gemm16_wmma(const float* __restrict__ A,
                            const float* __restrict__ B,
                            const float* __restrict__ bias,
                            float* __restrict__ Out,
                            const float* __restrict__ outw,
                            float* __restrict__ acc,
                            int M, int N) {
  const int wave = (int)((blockIdx.x * (long)blockDim.x + threadIdx.x) >> 5);
  const int lane = threadIdx.x & 31;
  const int ntiles = N >> 6;
  const int mt = wave / ntiles;
  const int nt = wave % ntiles;
  if (mt * 16 >= M) return;                 // wave-uniform: EXEC stays all-ones
  const int lr   = lane & 15;
  const int half = lane >> 4;
  const int row  = mt * 16 + lr;
  const int n0   = nt * 64 + lr;

  v8f c0 = {}; v8f c1 = {}; v8f c2 = {}; v8f c3 = {};

  // A 16x4 fragment: lanes 0-15 -> K={k,k+1}, lanes 16-31 -> K={k+2,k+3}
  const float* Ap = A + (size_t)row * LDA + 2 * half;   // 8B aligned (LDA even)
  const float* Bp = BT ? (B + (size_t)n0 * LDB + 2 * half)
                       : (B + (size_t)(2 * half) * LDB + n0);

  for (int k = 0; k < K; k += 4) {
    if ((k & 63) == 0) __builtin_prefetch(Ap + 64, 0, 1);
    v2f a = *(const v2f*)Ap;                 // single global_load_b64
    Ap += 4;
    v2f b0, b1, b2, b3;
    if (BT) {
      b0 = *(const v2f*)(Bp);
      b1 = *(const v2f*)(Bp + 16 * LDB);
      b2 = *(const v2f*)(Bp + 32 * LDB);
      b3 = *(const v2f*)(Bp + 48 * LDB);
      Bp += 4;
    } else {
      b0.x = Bp[0];  b0.y = Bp[LDB + 0];
      b1.x = Bp[16]; b1.y = Bp[LDB + 16];
      b2.x = Bp[32]; b2.y = Bp[LDB + 32];
      b3.x = Bp[48]; b3.y = Bp[LDB + 48];
      Bp += 4 * LDB;
    }
    c0 = __builtin_amdgcn_wmma_f32_16x16x4_f32(false, a, false, b0, (short)0, c0, false, false);
    c1 = __builtin_amdgcn_wmma_f32_16x16x4_f32(false, a, false, b1, (short)0, c1, false, false);
    c2 = __builtin_amdgcn_wmma_f32_16x16x4_f32(false, a, false, b2, (short)0, c2, false, false);
    c3 = __builtin_amdgcn_wmma_f32_16x16x4_f32(false, a, false, b3, (short)0, c3, false, false);
  }

  v8f cc[4] = {c0, c1, c2, c3};
  if (EPI == 0) {
    #pragma unroll
    for (int f = 0; f < 4; ++f) {
      int col = nt * 64 + f * 16 + lr;
      float bv = bias ? bias[col] : 0.0f;
      #pragma unroll
      for (int v = 0; v < 8; ++v) {
        int r = mt * 16 + v + 8 * half;     // D layout: VGPR v holds M = v + 8*half
        Out[(size_t)r * LDO + col] = cc[f][v] + bv;
      }
    }
  } else {
    float local = 0.0f;
    #pragma unroll
    for (int f = 0; f < 4; ++f) {
      int col = nt * 64 + f * 16 + lr;
      float s = 0.0f;
      #pragma unroll
      for (int v = 0; v < 8; ++v) s += cc[f][v];
      local += outw[col] * s;
    }
    #pragma unroll
    for (int off = 16; off > 0; off >>= 1) local += __shfl_down(local, off, 32);
    if (lane == 0) atomicAdd(acc, local);
  }
}

// ---------------------------------------------------------------- edge aggregate
// agg[dst] += coef * h[src] for both convs at once (h is [N,512]: s|f).
__global__ void aggregate_kernel(const int* __restrict__ src, const int* __restrict__ dst,
                                 const float* __restrict__ dis, const float* __restrict__ h,
                                 float* __restrict__ agg) {
  int wid  = (int)((blockIdx.x * (long)blockDim.x + threadIdx.x) >> 5);
  int lane = threadIdx.x & 31;
  if (wid >= N_EDGES) return;
  int s = src[wid], d = dst[wid];
  float coef = dis[s] * dis[d];
  const float* hs = h + (size_t)s * 512;
  float* ad = agg + (size_t)d * 512;
  #pragma unroll
  for (int i = 0; i < 512; i += 32)
    atomicAdd(&ad[i + lane], coef * hs[i + lane]);
}

// ---------------------------------------------------------------- combine (ReLU)
// comb = relu(agg + deg_inv * h + bias), float4-vectorized, in place over agg.
__global__ void combine_kernel(const float4* __restrict__ h4, float4* __restrict__ agg4,
                               const float* __restrict__ dis,
                               const float4* __restrict__ bs4, const float4* __restrict__ bf4) {
  long i = (long)blockIdx.x * blockDim.x + threadIdx.x;
  if (i >= (long)N_NODES * 128) return;     // 128 float4 per node (512 floats)
  int n = (int)(i >> 7);
  int j = (int)(i & 127);
  float4 bias = (j < 64) ? bs4[j] : bf4[j - 64];
  float di = dis[n] * dis[n];               // 1/deg
  float4 av = agg4[i], hv = h4[i], r;
  r.x = fmaxf(av.x + di * hv.x + bias.x, 0.0f);
  r.y = fmaxf(av.y + di * hv.y + bias.y, 0.0f);
  r.z = fmaxf(av.z + di * hv.z + bias.z, 0.0f);
  r.w = fmaxf(av.w + di * hv.w + bias.w, 0.0f);
  agg4[i] = r;
}

// ---------------------------------------------------------------- 2-token attention
// Per (node, head): 2x2 scores over 32-dim, softmax, o = attn @ v.
__global__ void attention_kernel(const float* __restrict__ qkv, float* __restrict__ o,
                                 int nodes) {
  int t = blockIdx.x * blockDim.x + threadIdx.x;
  if (t >= nodes * 8) return;
  int n = t >> 3, hd = t & 7;
  const float* r0 = qkv + (size_t)(2 * n) * 768;
  const float* r1 = r0 + 768;
  const float* q0 = r0 + hd * 32;       const float* q1 = r1 + hd * 32;
  const float* k0 = r0 + 256 + hd * 32; const float* k1 = r1 + 256 + hd * 32;
  const float* v0 = r0 + 512 + hd * 32; const float* v1 = r1 + 512 + hd * 32;
  float s00 = 0, s01 = 0, s10 = 0, s11 = 0;
  #pragma unroll 8
  for (int d = 0; d < 32; ++d) {
    float a = q0[d], b = q1[d], c = k0[d], e = k1[d];
    s00 += a * c; s01 += a * e; s10 += b * c; s11 += b * e;
  }
  const float scale = 0.17677669529663687f;   // 1/sqrt(32)
  s00 *= scale; s01 *= scale; s10 *= scale; s11 *= scale;
  float m0 = fmaxf(s00, s01), m1 = fmaxf(s10, s11);
  float e00 = expf(s00 - m0), e01 = expf(s01 - m0);
  float e10 = expf(s10 - m1), e11 = expf(s11 - m1);
  float z0 = e00 + e01, z1 = e10 + e11;
  float a00 = e00 / z0, a01 = e01 / z0, a10 = e10 / z1, a11 = e11 / z1;
  float* o0 = o + (size_t)(2 * n) * 256 + hd * 32;
  float* o1 = o0 + 256;
  #pragma unroll 8
  for (int d = 0; d < 32; ++d) {
    float vv0 = v0[d], vv1 = v1[d];
    o0[d] = a00 * vv0 + a01 * vv1;
    o1[d] = a10 * vv0 + a11 * vv1;
  }
}

// ---------------------------------------------------------------- final scalar
// out = acc/(2N) + dot(out_proj_b, out_w) + out_b   (bias folded out of GEMM)
__global__ void final_kernel(const float* __restrict__ acc, const float* __restrict__ opb,
                             const float* __restrict__ ow, const float* __restrict__ ob,
                             float* __restrict__ out) {
  __shared__ float red[256];
  int t = threadIdx.x;
  red[t] = opb[t] * ow[t];
  __syncthreads();
  for (int s = 128; s > 0; s >>= 1) {
    if (t < s) red[t] += red[t + s];
    __syncthreads();
  }
  if (t == 0) out[0] = acc[0] * (1.0f / (2.0f * N_NODES)) + red[0] + ob[0];
}

// ---------------------------------------------------------------- launch
extern "C" void kernel_launch(void* const* d_in, const int* in_sizes, int n_in,
                              void* d_out, int out_size, void* d_ws, size_t ws_size,
                              hipStream_t stream) {
  (void)in_sizes; (void)n_in; (void)out_size; (void)ws_size;
  const float* x   = (const float*)d_in[0];
  const int*   ei  = (const int*)  d_in[1];
  const float* W_s = (const float*)d_in[2];
  const float* b_s = (const float*)d_in[3];
  const float* W_f = (const float*)d_in[4];
  const float* b_f = (const float*)d_in[5];
  const float* ipw = (const float*)d_in[6];
  const float* ipb = (const float*)d_in[7];
  const float* opw = (const float*)d_in[8];
  const float* opb = (const float*)d_in[9];
  const float* ow  = (const float*)d_in[10];
  const float* ob  = (const float*)d_in[11];
  float* out = (float*)d_out;

  const int* src = ei;
  const int* dst = ei + N_EDGES;

  // workspace layout (floats): ~236 MB total; all segments 16B-aligned
  float* ws   = (float*)d_ws;
  float* deg  = ws;                             // N
  float* dis  = deg + N_NODES;                  // N
  float* acc  = dis + N_NODES;                  // 8 (pad)
  float* h    = acc + 8;                        // N*512 (later reused as o_chunk)
  float* agg  = h + (size_t)N_NODES * 512;      // N*512 (becomes comb [2N,256])
  float* qkvc = agg + (size_t)N_NODES * 512;    // 10000*768 per chunk

  // 1) init scratch (float4 stores)
  {
    long total4 = (long)N_NODES * 128;
    init_ws_kernel<<<(int)((total4 + 255) / 256), 256, 0, stream>>>(
        (float4*)agg, deg, acc, total4);
  }
  // 2) degrees + normalization
  degree_kernel<<<(N_EDGES + 255) / 256, 256, 0, stream>>>(dst, deg);
  rsqrt_kernel<<<(N_NODES + 255) / 256, 256, 0, stream>>>(deg, dis);

  // 3) feature GEMMs: h[:, :256] = x_s @ W_s, h[:, 256:] = x_f @ W_f
  {
    int waves  = (N_NODES / 16) * (256 / 64);
    int blocks = (waves * 32 + 255) / 256;
    gemm16_wmma<false, 0, 300, 256, 512, 100><<<blocks, 256, 0, stream>>>(
        x, W_s, nullptr, h, nullptr, nullptr, N_NODES, 256);
    gemm16_wmma<false, 0, 300, 256, 512, 200><<<blocks, 256, 0, stream>>>(
        x + 100, W_f, nullptr, h + 256, nullptr, nullptr, N_NODES, 256);
  }

  // 4) edge aggregation (one wave per edge; atomics land in L2-resident agg)
  aggregate_kernel<<<(N_EDGES * 32 + 255) / 256, 256, 0, stream>>>(src, dst, dis, h, agg);

  // 5) combine + ReLU -> comb [2N, 256] (in place over agg)
  {
    long total4 = (long)N_NODES * 128;
    combine_kernel<<<(int)((total4 + 255) / 256), 256, 0, stream>>>(
        (const float4*)h, (float4*)agg, dis, (const float4*)b_s, (const float4*)b_f);
  }

  // 6) chunked attention head: qkv GEMM -> attention -> out_proj GEMM w/ fused reduce
  const int NCH = 10, NODES_C = N_NODES / NCH;  // 5000 nodes -> 10000 rows, %16 == 0
  for (int c = 0; c < NCH; ++c) {
    const float* combA = agg + (size_t)c * NODES_C * 512;  // rows [2c*NODES_C, ...)
    const int Mrows = 2 * NODES_C;                         // 10000
    {
      int waves  = (Mrows / 16) * (768 / 64);
      int blocks = (waves * 32 + 255) / 256;
      gemm16_wmma<true, 0, 256, 256, 768, 256><<<blocks, 256, 0, stream>>>(
          combA, ipw, ipb, qkvc, nullptr, nullptr, Mrows, 768);
    }
    attention_kernel<<<(NODES_C * 8 + 255) / 256, 256, 0, stream>>>(qkvc, h, NODES_C);
    {
      int waves  = (Mrows / 16) * (256 / 64);
      int blocks = (waves * 32 + 255) / 256;
      gemm16_wmma<true, 1, 256, 256, 256, 256><<<blocks, 256, 0, stream>>>(
          h, opw, nullptr, nullptr, ow, acc, Mrows, 256);
    }
  }

  // 7) final scalar
  final_kernel<<<1, 256, 0, stream>>>(acc, opb, ow, ob, out);
}